// Shm_37529424233278
// MI455X (gfx1250) — compile-verified
//
#include <hip/hip_runtime.h>

#ifndef __has_builtin
#define __has_builtin(x) 0
#endif

#define T_DIM 1460
#define B_DIM 2048
#define NMUL  4
#define LENF  15
#define HIST  (LENF - 1)   // 14 history rows for the causal 15-tap filter
#define TT    64           // time rows per conv tile
#define BBAS  64           // basins per conv tile
#define TB    ((size_t)T_DIM * (size_t)B_DIM)

typedef __attribute__((ext_vector_type(4))) unsigned int tdm_g0_t;
typedef __attribute__((ext_vector_type(8))) int          tdm_g1_t;
typedef __attribute__((ext_vector_type(4))) int          tdm_g2_t;

// ---------------------------------------------------------------------------
// TDM: 2-D tile load (row-major tensor, stride0 elements between rows) -> LDS.
// D# packing per CDNA5 ISA §8.3 (group 0) / §8.4 (group 1); data_size = 4B.
// Called UNCONDITIONALLY: a clean compile proves the builtin exists+lowers.
// ---------------------------------------------------------------------------
__device__ __forceinline__ void tdm_load_tile_2d(unsigned lds_addr,
                                                 unsigned long long gaddr,
                                                 unsigned tile_d0, unsigned tile_d1,
                                                 unsigned tensor_d0, unsigned tensor_d1,
                                                 unsigned long long stride0_elems) {
    tdm_g0_t g0;
    g0[0] = 1u;                                                // count=1, user D#
    g0[1] = lds_addr;                                          // lds_addr[31:0]
    g0[2] = (unsigned)(gaddr & 0xFFFFFFFFull);                 // global_addr lo
    g0[3] = (unsigned)((gaddr >> 32) & 0x1FFFFFFull)           // global_addr[56:32]
          | (2u << 30);                                        // type = 2 ("image")
    tdm_g1_t g1;
    g1[0] = (int)(2u << 16);                                   // wg_mask=0, data_size=2 (4B)
    g1[1] = (int)((tensor_d0 & 0xFFFFu) << 16);                // tensor_dim0[15:0] @ bits63:48
    g1[2] = (int)(((tensor_d0 >> 16) & 0xFFFFu)                // tensor_dim0[31:16]
          | ((tensor_d1 & 0xFFFFu) << 16));                    // tensor_dim1[15:0]
    g1[3] = (int)(((tensor_d1 >> 16) & 0xFFFFu)                // tensor_dim1[31:16]
          | ((tile_d0 & 0xFFFFu) << 16));                      // tile_dim0
    g1[4] = (int)(tile_d1 & 0xFFFFu);                          // tile_dim1 (tile_dim2=0)
    g1[5] = (int)(stride0_elems & 0xFFFFFFFFull);              // tensor_dim0_stride lo32
    g1[6] = (int)((stride0_elems >> 32) & 0xFFFFull);          // stride hi16 (dim1_stride=0)
    g1[7] = 0;
    tdm_g2_t z4 = {0, 0, 0, 0};
#if defined(__clang_major__) && (__clang_major__ >= 23)
    tdm_g1_t z8 = {0, 0, 0, 0, 0, 0, 0, 0};
    __builtin_amdgcn_tensor_load_to_lds(g0, g1, z4, z4, z8, 0);
#else
    __builtin_amdgcn_tensor_load_to_lds(g0, g1, z4, z4, 0);
#endif
}

__device__ __forceinline__ void wait_tensorcnt0() {
#if __has_builtin(__builtin_amdgcn_s_wait_tensorcnt)
    __builtin_amdgcn_s_wait_tensorcnt(0);
#else
    asm volatile("s_wait_tensorcnt 0x0" ::: "memory");
#endif
}

// ---------------------------------------------------------------------------
// Kernel 1: gamma unit-hydrograph weights, one thread per basin.
// The 1/(Gamma(a) * theta^a) normalizer cancels under renormalization.
// w layout: [LENF, B] row-major (matches TDM dim0-stride = B).
// ---------------------------------------------------------------------------
__global__ void shm_uh_kernel(const float* __restrict__ ps, float* __restrict__ w) {
    const int b = blockIdx.x * blockDim.x + threadIdx.x;
    if (b >= B_DIM) return;
    const float ra = ps[b * 34 + 32] * 2.9f;            // [0, 2.9]
    const float rb = ps[b * 34 + 33] * 6.5f;            // [0, 6.5]
    const float aa = fmaxf(ra, 0.0f) + 0.1f;
    const float th = fmaxf(rb, 0.0f) + 0.5f;
    float vals[LENF];
    float s = 0.0f;
#pragma unroll
    for (int k = 0; k < LENF; ++k) {
        const float t = (float)k + 0.5f;
        const float v = __powf(t, aa - 1.0f) * __expf(-t / th);
        vals[k] = v;
        s += v;
    }
    const float inv = 1.0f / s;
#pragma unroll
    for (int k = 0; k < LENF; ++k) w[k * B_DIM + b] = vals[k] * inv;
}

// quad (4-lane) mean via wave32 xor-shuffles
__device__ __forceinline__ float quad_mean(float v) {
    v += __shfl_xor(v, 1, 32);
    v += __shfl_xor(v, 2, 32);
    return v * 0.25f;
}

// ---------------------------------------------------------------------------
// Kernel 2: the T=1460 recurrence. One lane per (basin, ensemble-member);
// lanes [4q..4q+3] share a basin, ensemble mean via quad shuffles.
// Prefetches forcing 8 steps ahead (global_prefetch_b8) to hide HBM latency
// along the serial dependence chain.
// ---------------------------------------------------------------------------
__global__ void shm_scan_kernel(const float* __restrict__ x,
                                const float* __restrict__ ps,
                                float* __restrict__ qplanes,   // 4 planes of [T,B]
                                float* __restrict__ out) {
    const int tid = blockIdx.x * blockDim.x + threadIdx.x;     // 0 .. B*NMUL-1
    const int b = tid >> 2;
    const int m = tid & 3;
    if (b >= B_DIM) return;

    const float lo[8] = {0.0f, 10.0f, 20.0f, 1.0f, 0.0f, 1.0f, 1.0f, 10.0f};
    const float hi[8] = {10.0f, 60.0f, 700.0f, 6.0f, 1.0f, 20.0f, 100.0f, 1000.0f};
    float prm[8];
#pragma unroll
    for (int i = 0; i < 8; ++i)
        prm[i] = ps[b * 34 + i * NMUL + m] * (hi[i] - lo[i]) + lo[i];
    const float dd = prm[0], f_thr = prm[1], sumax = prm[2], beta = prm[3];
    const float perc = prm[4], kf = prm[5], ki = prm[6], kb = prm[7];
    const float inv_sumax = 1.0f / sumax;

    float ss = 0.0f, sf = 1.0f, su = 5.0f, si = 10.0f, sb = 15.0f;

    for (int t = 0; t < T_DIM; ++t) {
        if (t + 8 < T_DIM)   // speculative RT prefetch of forcing 8 steps ahead
            __builtin_prefetch(x + ((size_t)(t + 8) * B_DIM + b) * 4, 0, 0);
        const float4 f = *(const float4*)(x + ((size_t)t * B_DIM + b) * 4);
        const float p = f.x, tmn = f.y, tmx = f.z, pet = f.w;
        const float tmean = 0.5f * (tmn + tmx);
        // snow
        const float snowfall = (tmean < 0.0f) ? p : 0.0f;
        const float rain = p - snowfall;
        ss += snowfall;
        const float melt = fminf(ss, fmaxf(dd * tmean, 0.0f));
        ss -= melt;
        const float qin = rain + melt;
        // fast flow
        const float excess = fmaxf(qin - f_thr, 0.0f);
        sf += excess;
        const float qf = sf / kf;
        sf -= qf;
        const float remain = qin - excess;
        // unsaturated zone
        const float r0 = fminf(fmaxf(su * inv_sumax, 0.0f), 1.0f);
        const float psi = __powf(r0, beta);
        su += (1.0f - psi) * remain;
        const float r1 = fminf(fmaxf(su * inv_sumax, 0.0f), 1.0f);
        const float aet = fminf(pet * r1, su);
        su -= aet;
        const float qu = psi * remain;
        // interflow / baseflow
        si += (1.0f - perc) * qu;
        const float qi = si / ki;
        si -= qi;
        sb += perc * qu;
        const float qb = sb / kb;
        sb -= qb;

        const float qs_m  = quad_mean(qf + qi + qb);
        const float qf_m  = quad_mean(qf);
        const float qi_m  = quad_mean(qi);
        const float qb_m  = quad_mean(qb);
        const float aet_m = quad_mean(aet);
        const float ss_m  = quad_mean(ss);
        if (m == 0) {
            const size_t o = (size_t)t * B_DIM + b;
            qplanes[o]          = qs_m;
            qplanes[TB + o]     = qf_m;
            qplanes[2 * TB + o] = qi_m;
            qplanes[3 * TB + o] = qb_m;
            out[4 * TB + o] = aet_m;
            out[5 * TB + o] = ss_m;
        }
    }
}

// ---------------------------------------------------------------------------
// Kernel 3: per-basin causal 15-tap conv. Tiles [TT x BBAS]; the q history
// tile ([TT+14, 64]) and the filter tile ([15, 64]) are DMA'd into LDS by
// the Tensor Data Mover (wave 0 issues; TDM ignores EXEC; TENSORcnt waits).
// grid.z = signal index (qsim, qf, qi, qb).
// ---------------------------------------------------------------------------
__global__ void shm_conv_kernel(const float* __restrict__ qplanes,
                                const float* __restrict__ w,
                                float* __restrict__ out) {
    __shared__ float qtile[(TT + HIST) * BBAS];
    __shared__ float wtile[LENF * BBAS];

    const int t0  = blockIdx.x * TT;
    const int b0  = blockIdx.y * BBAS;
    const int sig = blockIdx.z;
    const float* qsig = qplanes + (size_t)sig * TB;

    const int row_start   = (t0 >= HIST) ? (t0 - HIST) : 0;
    const int lds_row_off = (t0 >= HIST) ? 0 : (HIST - t0);
    int t_end = t0 + TT;
    if (t_end > T_DIM) t_end = T_DIM;
    const int rows = t_end - row_start;

    // zero the pre-history rows (t < 0) for the first time-tile
    if (lds_row_off > 0) {
        for (int i = threadIdx.x; i < lds_row_off * BBAS; i += blockDim.x)
            qtile[i] = 0.0f;
    }

    if (threadIdx.x < 32) {   // wave 0 only: TDM ignores EXEC, issue once
        const unsigned lds_q = (unsigned)(size_t)(qtile + lds_row_off * BBAS);
        const unsigned long long ga_q =
            (unsigned long long)(size_t)(qsig + (size_t)row_start * B_DIM + b0);
        tdm_load_tile_2d(lds_q, ga_q,
                         /*tile*/ BBAS, (unsigned)rows,
                         /*tensor remaining*/ (unsigned)(B_DIM - b0),
                         (unsigned)(T_DIM - row_start),
                         /*stride0*/ (unsigned long long)B_DIM);
        const unsigned lds_w = (unsigned)(size_t)(wtile);
        const unsigned long long ga_w = (unsigned long long)(size_t)(w + b0);
        tdm_load_tile_2d(lds_w, ga_w,
                         BBAS, LENF,
                         (unsigned)(B_DIM - b0), LENF,
                         (unsigned long long)B_DIM);
        wait_tensorcnt0();
    }
    __syncthreads();

    const int tx = threadIdx.x & (BBAS - 1);   // basin within tile
    const int ty = threadIdx.x >> 6;           // 0..3, covers 16 time rows each
    for (int tr = ty; tr < TT; tr += 4) {
        const int t = t0 + tr;
        if (t >= T_DIM) break;
        float acc = 0.0f;
#pragma unroll
        for (int k = 0; k < LENF; ++k)
            acc = fmaf(wtile[k * BBAS + tx], qtile[(tr + HIST - k) * BBAS + tx], acc);
        out[(size_t)sig * TB + (size_t)t * B_DIM + (b0 + tx)] = acc;
    }
}

// ---------------------------------------------------------------------------
extern "C" void kernel_launch(void* const* d_in, const int* in_sizes, int n_in,
                              void* d_out, int out_size, void* d_ws, size_t ws_size,
                              hipStream_t stream) {
    const float* x_phy = (const float*)d_in[0];   // [T,B,4]
    const float* ps    = (const float*)d_in[1];   // [B,34]
    float* out = (float*)d_out;                   // [6,T,B]
    float* ws  = (float*)d_ws;

    float* qplanes = ws;                 // 4 * T*B floats (qsim,qf,qi,qb means)
    float* wuh     = ws + 4 * TB;        // LENF * B floats
    (void)in_sizes; (void)n_in; (void)out_size; (void)ws_size;

    shm_uh_kernel<<<(B_DIM + 255) / 256, 256, 0, stream>>>(ps, wuh);
    shm_scan_kernel<<<(B_DIM * NMUL + 255) / 256, 256, 0, stream>>>(x_phy, ps, qplanes, out);
    dim3 grid((T_DIM + TT - 1) / TT, B_DIM / BBAS, 4);
    shm_conv_kernel<<<grid, 256, 0, stream>>>(qplanes, wuh, out);
}